// AttentionLayerPreNorm_3985729650927
// MI455X (gfx1250) — compile-verified
//
#include <hip/hip_runtime.h>
#include <stdint.h>
#include <stddef.h>

// ---------------------------------------------------------------------------
// Transformer layer (pre-norm MHA + GShard top-2 MoE), fp32, for MI455X
// (gfx1250). All matmuls use V_WMMA_F32_16X16X4_F32 (fp32-exact matrix path).
// ---------------------------------------------------------------------------

#define L_DIM   2048
#define B_DIM   4
#define E_DIM   512
#define H_DIM   8
#define HD      64
#define FF_DIM  2048
#define NE_DIM  5
#define CAP     4
#define QKV_DIM (3 * E_DIM)
#define TOKENS  (L_DIM * B_DIM)   // 8192

typedef __attribute__((ext_vector_type(2))) float v2f;
typedef __attribute__((ext_vector_type(8))) float v8f;

__device__ __forceinline__ v8f wmma_f32(v2f a, v2f b, v8f c) {
  // D = A(16x4,f32) * B(4x16,f32) + C(16x16,f32)
  return __builtin_amdgcn_wmma_f32_16x16x4_f32(false, a, false, b, (short)0, c,
                                               false, false);
}

__device__ __forceinline__ v8f v8f_zero() {
  v8f z = {0.f, 0.f, 0.f, 0.f, 0.f, 0.f, 0.f, 0.f};
  return z;
}

// ---------------------------------------------------------------------------
// LayerNorm: one wave per row of E_DIM=512.
// ---------------------------------------------------------------------------
__global__ void __launch_bounds__(256)
ln_kernel(const float* __restrict__ x, const float* __restrict__ w,
          const float* __restrict__ b, float* __restrict__ y, int rows) {
  const int lane = threadIdx.x & 31;
  const int wave = threadIdx.x >> 5;
  const int row = blockIdx.x * 8 + wave;
  if (row >= rows) return;
  const float* xr = x + (size_t)row * E_DIM;
  float v[16];
  float s = 0.f;
#pragma unroll
  for (int i = 0; i < 16; ++i) { v[i] = xr[lane + 32 * i]; s += v[i]; }
#pragma unroll
  for (int m = 16; m >= 1; m >>= 1) s += __shfl_xor(s, m, 32);
  const float mu = s * (1.f / E_DIM);
  float var = 0.f;
#pragma unroll
  for (int i = 0; i < 16; ++i) { float d = v[i] - mu; var += d * d; }
#pragma unroll
  for (int m = 16; m >= 1; m >>= 1) var += __shfl_xor(var, m, 32);
  const float inv = rsqrtf(var * (1.f / E_DIM) + 1e-5f);
  float* yr = y + (size_t)row * E_DIM;
#pragma unroll
  for (int i = 0; i < 16; ++i) {
    int c = lane + 32 * i;
    yr[c] = (v[i] - mu) * inv * w[c] + b[c];
  }
}

// ---------------------------------------------------------------------------
// GEMM  C = A(MxK) @ Bw(NxK)^T + bias (+ residual).
// One wave per 16x64 strip: the A fragment is loaded once per k-step and
// reused across 4 WMMAs (5 b64 loads / 4 WMMAs instead of 8 / 4).
// ---------------------------------------------------------------------------
__global__ void __launch_bounds__(256)
gemm_tn_wmma(const float* __restrict__ A, const float* __restrict__ Bw,
             const float* __restrict__ bias, const float* __restrict__ resid,
             float* __restrict__ C, int M, int N, int K) {
  const int lane = threadIdx.x & 31;
  const int wave = threadIdx.x >> 5;
  const int lm = lane & 15;
  const int hi = lane >> 4;
  const int tilesN = N >> 6;  // 64-wide strips
  const int tile = blockIdx.x * 8 + wave;
  if (tile >= (M >> 4) * tilesN) return;
  const int m0 = (tile / tilesN) << 4;
  const int n0 = (tile % tilesN) << 6;

  const float* arow = A + (size_t)(m0 + lm) * K;
  const float* brow0 = Bw + (size_t)(n0 + 0 * 16 + lm) * K;
  const float* brow1 = Bw + (size_t)(n0 + 1 * 16 + lm) * K;
  const float* brow2 = Bw + (size_t)(n0 + 2 * 16 + lm) * K;
  const float* brow3 = Bw + (size_t)(n0 + 3 * 16 + lm) * K;

  // Warm the rows this wave will stream (global_prefetch_b8).
  __builtin_prefetch(arow, 0, 1);
  __builtin_prefetch(brow0, 0, 1);
  __builtin_prefetch(brow1, 0, 1);
  __builtin_prefetch(brow2, 0, 1);
  __builtin_prefetch(brow3, 0, 1);

  v8f acc[4];
#pragma unroll
  for (int i = 0; i < 4; ++i) acc[i] = v8f_zero();

  for (int k = 0; k < K; k += 4) {
    const int ko = k + 2 * hi;
    v2f a = *(const v2f*)(arow + ko);
    v2f b0 = *(const v2f*)(brow0 + ko);
    v2f b1 = *(const v2f*)(brow1 + ko);
    v2f b2 = *(const v2f*)(brow2 + ko);
    v2f b3 = *(const v2f*)(brow3 + ko);
    acc[0] = wmma_f32(a, b0, acc[0]);
    acc[1] = wmma_f32(a, b1, acc[1]);
    acc[2] = wmma_f32(a, b2, acc[2]);
    acc[3] = wmma_f32(a, b3, acc[3]);
  }

#pragma unroll
  for (int nt = 0; nt < 4; ++nt) {
    const int col = n0 + nt * 16 + lm;
    const float bv = bias ? bias[col] : 0.f;
#pragma unroll
    for (int r = 0; r < 8; ++r) {
      const int row = m0 + r + 8 * hi;
      float v = acc[nt][r] + bv;
      if (resid) v += resid[(size_t)row * N + col];
      C[(size_t)row * N + col] = v;
    }
  }
}

// ---------------------------------------------------------------------------
// Flash-style attention. One wave per (b, h, 16-query tile). Online softmax.
// qkv layout: (L, B, 3E) with q|k|v at offsets 0|E|2E, head h at h*HD.
// ---------------------------------------------------------------------------
__global__ void __launch_bounds__(32)
attn_kernel(const float* __restrict__ qkv, float* __restrict__ out) {
  const int lane = threadIdx.x & 31;
  const int lm = lane & 15;
  const int hi = lane >> 4;
  const int qt = blockIdx.x & 127;
  const int bh = blockIdx.x >> 7;
  const int b = bh & (B_DIM - 1);
  const int h = bh >> 2;
  const int q0 = qt << 4;

  __shared__ float pbuf[16][17];

  v2f aq[16];
  {
    const float* qrow =
        qkv + ((size_t)(q0 + lm) * B_DIM + b) * QKV_DIM + h * HD;
#pragma unroll
    for (int j = 0; j < 16; ++j) {
      v2f t = *(const v2f*)(qrow + 4 * j + 2 * hi);
      aq[j].x = t.x * 0.125f;  // hd^-0.5 = 1/8
      aq[j].y = t.y * 0.125f;
    }
  }

  v8f o[4];
#pragma unroll
  for (int i = 0; i < 4; ++i) o[i] = v8f_zero();
  float mrun[8], lrun[8];
#pragma unroll
  for (int r = 0; r < 8; ++r) { mrun[r] = -3.4e38f; lrun[r] = 0.f; }

  for (int t = 0; t < L_DIM / 16; ++t) {
    // S = Q K^T  (16x16 over 64-deep head dim)
    v8f s = v8f_zero();
    const float* krow =
        qkv + ((size_t)(t * 16 + lm) * B_DIM + b) * QKV_DIM + E_DIM + h * HD;
#pragma unroll
    for (int j = 0; j < 16; ++j) {
      v2f kb = *(const v2f*)(krow + 4 * j + 2 * hi);
      s = wmma_f32(aq[j], kb, s);
    }
    // online softmax per row (row = r + 8*hi; N striped over 16-lane half)
#pragma unroll
    for (int r = 0; r < 8; ++r) {
      float v = s[r];
      v = fmaxf(v, __shfl_xor(v, 1, 16));
      v = fmaxf(v, __shfl_xor(v, 2, 16));
      v = fmaxf(v, __shfl_xor(v, 4, 16));
      v = fmaxf(v, __shfl_xor(v, 8, 16));
      const float nm = fmaxf(mrun[r], v);
      const float corr = __expf(mrun[r] - nm);
      const float p = __expf(s[r] - nm);
      float ps = p;
      ps += __shfl_xor(ps, 1, 16);
      ps += __shfl_xor(ps, 2, 16);
      ps += __shfl_xor(ps, 4, 16);
      ps += __shfl_xor(ps, 8, 16);
      lrun[r] = lrun[r] * corr + ps;
      mrun[r] = nm;
      s[r] = p;
      o[0][r] *= corr; o[1][r] *= corr; o[2][r] *= corr; o[3][r] *= corr;
    }
    // reshape P: C-layout -> A-layout through LDS
    __syncthreads();
#pragma unroll
    for (int r = 0; r < 8; ++r) pbuf[r + 8 * hi][lm] = s[r];
    __syncthreads();
    v2f ap[4];
#pragma unroll
    for (int j = 0; j < 4; ++j) {
      ap[j].x = pbuf[lm][4 * j + 2 * hi];
      ap[j].y = pbuf[lm][4 * j + 2 * hi + 1];
    }
    // O += P V
#pragma unroll
    for (int j = 0; j < 4; ++j) {
      const float* v0 = qkv +
          ((size_t)(t * 16 + 4 * j + 2 * hi) * B_DIM + b) * QKV_DIM +
          2 * E_DIM + h * HD;
      const float* v1 = v0 + (size_t)B_DIM * QKV_DIM;
#pragma unroll
      for (int nt = 0; nt < 4; ++nt) {
        v2f vb;
        vb.x = v0[nt * 16 + lm];
        vb.y = v1[nt * 16 + lm];
        o[nt] = wmma_f32(ap[j], vb, o[nt]);
      }
    }
  }
#pragma unroll
  for (int nt = 0; nt < 4; ++nt)
#pragma unroll
    for (int r = 0; r < 8; ++r) {
      const int row = q0 + r + 8 * hi;
      out[((size_t)row * B_DIM + b) * E_DIM + h * HD + nt * 16 + lm] =
          o[nt][r] / lrun[r];
    }
}

// ---------------------------------------------------------------------------
// Threefry-2x32 (JAX PRNG), 20 rounds.
// ---------------------------------------------------------------------------
__device__ __forceinline__ unsigned rotl32(unsigned x, int r) {
  return (x << r) | (x >> (32 - r));
}
__device__ __forceinline__ void threefry2x32(unsigned k0, unsigned k1,
                                             unsigned c0, unsigned c1,
                                             unsigned& o0, unsigned& o1) {
  const unsigned ks2 = k0 ^ k1 ^ 0x1BD11BDAu;
  unsigned x0 = c0 + k0, x1 = c1 + k1;
  const int ra[4] = {13, 15, 26, 6};
  const int rb[4] = {17, 29, 16, 24};
#pragma unroll
  for (int i = 0; i < 4; ++i) { x0 += x1; x1 = rotl32(x1, ra[i]); x1 ^= x0; }
  x0 += k1; x1 += ks2 + 1u;
#pragma unroll
  for (int i = 0; i < 4; ++i) { x0 += x1; x1 = rotl32(x1, rb[i]); x1 ^= x0; }
  x0 += ks2; x1 += k0 + 2u;
#pragma unroll
  for (int i = 0; i < 4; ++i) { x0 += x1; x1 = rotl32(x1, ra[i]); x1 ^= x0; }
  x0 += k0; x1 += k1 + 3u;
#pragma unroll
  for (int i = 0; i < 4; ++i) { x0 += x1; x1 = rotl32(x1, rb[i]); x1 ^= x0; }
  x0 += k1; x1 += ks2 + 4u;
#pragma unroll
  for (int i = 0; i < 4; ++i) { x0 += x1; x1 = rotl32(x1, ra[i]); x1 ^= x0; }
  x0 += ks2; x1 += k0 + 5u;
  o0 = x0; o1 = x1;
}

// ---------------------------------------------------------------------------
// Top-2 gating. One wave per group b (L groups of B tokens).
// Writes: slot_tok/slot_gate [NE][L][CAP], and per-token (e,c,g) x2.
// ---------------------------------------------------------------------------
__global__ void __launch_bounds__(32)
gating_kernel(const float* __restrict__ h2, const float* __restrict__ wg,
              int* __restrict__ slot_tok, float* __restrict__ slot_gate,
              int* __restrict__ tok_e, int* __restrict__ tok_c,
              float* __restrict__ tok_g) {
  const int lane = threadIdx.x;
  const int b = blockIdx.x;
  float lg[B_DIM][NE_DIM];
#pragma unroll
  for (int n = 0; n < B_DIM; ++n)
#pragma unroll
    for (int e = 0; e < NE_DIM; ++e) lg[n][e] = 0.f;
  for (int d = lane; d < E_DIM; d += 32) {
    float w[NE_DIM];
#pragma unroll
    for (int e = 0; e < NE_DIM; ++e) w[e] = wg[(size_t)d * NE_DIM + e];
#pragma unroll
    for (int n = 0; n < B_DIM; ++n) {
      const float xv = h2[((size_t)b * B_DIM + n) * E_DIM + d];
#pragma unroll
      for (int e = 0; e < NE_DIM; ++e) lg[n][e] += xv * w[e];
    }
  }
#pragma unroll
  for (int n = 0; n < B_DIM; ++n)
#pragma unroll
    for (int e = 0; e < NE_DIM; ++e) {
      float v = lg[n][e];
      v += __shfl_xor(v, 16, 32);
      v += __shfl_xor(v, 8, 32);
      v += __shfl_xor(v, 4, 32);
      v += __shfl_xor(v, 2, 32);
      v += __shfl_xor(v, 1, 32);
      lg[n][e] = v;
    }

  if (lane == 0) {
    int i1[B_DIM], i2[B_DIM], keep2[B_DIM];
    float g1v[B_DIM], g2v[B_DIM];
    for (int n = 0; n < B_DIM; ++n) {
      float mx = lg[n][0];
      for (int e = 1; e < NE_DIM; ++e) mx = fmaxf(mx, lg[n][e]);
      float p[NE_DIM], se = 0.f;
      for (int e = 0; e < NE_DIM; ++e) { p[e] = __expf(lg[n][e] - mx); se += p[e]; }
      for (int e = 0; e < NE_DIM; ++e) p[e] /= se;
      int a1 = 0;
      for (int e = 1; e < NE_DIM; ++e) if (p[e] > p[a1]) a1 = e;
      float q[NE_DIM];
      for (int e = 0; e < NE_DIM; ++e) q[e] = (e == a1) ? 0.f : p[e];
      int a2 = 0;
      for (int e = 1; e < NE_DIM; ++e) if (q[e] > q[a2]) a2 = e;
      float ga = p[a1], gb = q[a2];
      const float den = ga + gb + 1e-9f;
      ga /= den; gb /= den;
      // JAX uniform(key=42) at flat index li (count split at n/2=4096)
      const int li = b * B_DIM + n;
      unsigned c0, c1, r0, r1;
      if (li < TOKENS / 2) { c0 = (unsigned)li; c1 = (unsigned)(li + TOKENS / 2); }
      else                 { c0 = (unsigned)(li - TOKENS / 2); c1 = (unsigned)li; }
      threefry2x32(0u, 42u, c0, c1, r0, r1);
      const unsigned bits = (li < TOKENS / 2) ? r0 : r1;
      const float u = __uint_as_float((bits >> 9) | 0x3f800000u) - 1.0f;
      keep2[n] = (u < gb / 0.2f) ? 1 : 0;
      i1[n] = a1; i2[n] = a2; g1v[n] = ga; g2v[n] = gb;
    }
    // capacity bookkeeping (exclusive cumsum over tokens)
    int c1cnt[NE_DIM] = {0, 0, 0, 0, 0}, mcount[NE_DIM] = {0, 0, 0, 0, 0};
    int pos1[B_DIM], kept1[B_DIM];
    for (int n = 0; n < B_DIM; ++n) pos1[n] = c1cnt[i1[n]]++;
    for (int n = 0; n < B_DIM; ++n) {
      kept1[n] = (pos1[n] < CAP);
      if (kept1[n]) mcount[i1[n]]++;
    }
    int c2cnt[NE_DIM] = {0, 0, 0, 0, 0};
    int pos2[B_DIM], kept2[B_DIM];
    for (int n = 0; n < B_DIM; ++n) {
      if (keep2[n]) {
        pos2[n] = mcount[i2[n]] + c2cnt[i2[n]]++;
        kept2[n] = (pos2[n] < CAP);
      } else { pos2[n] = 0; kept2[n] = 0; }
    }
    // emit slot map + per-token assignments
    for (int e = 0; e < NE_DIM; ++e)
      for (int c = 0; c < CAP; ++c) {
        slot_tok[((size_t)e * L_DIM + b) * CAP + c] = -1;
        slot_gate[((size_t)e * L_DIM + b) * CAP + c] = 0.f;
      }
    for (int n = 0; n < B_DIM; ++n) {
      const int t = b * B_DIM + n;
      int e0 = 0, cc0 = 0; float gg0 = 0.f;
      if (kept1[n]) {
        e0 = i1[n]; cc0 = pos1[n]; gg0 = g1v[n];
        slot_tok[((size_t)e0 * L_DIM + b) * CAP + cc0] = n;
        slot_gate[((size_t)e0 * L_DIM + b) * CAP + cc0] = gg0;
      }
      tok_e[t] = e0; tok_c[t] = cc0; tok_g[t] = gg0;
      int e1 = 0, cc1 = 0; float gg1 = 0.f;
      if (kept2[n]) {
        e1 = i2[n]; cc1 = pos2[n]; gg1 = g2v[n];
        slot_tok[((size_t)e1 * L_DIM + b) * CAP + cc1] = n;
        slot_gate[((size_t)e1 * L_DIM + b) * CAP + cc1] = gg1;
      }
      tok_e[TOKENS + t] = e1; tok_c[TOKENS + t] = cc1; tok_g[TOKENS + t] = gg1;
    }
  }
}

// ---------------------------------------------------------------------------
// Fused MoE FFN: block = (expert e, 4 groups) -> 16 slot-rows.
// A (16x512) gathered to LDS; hidden processed in 256-wide chunks:
// GEMM1 -> ReLU (LDS) -> GEMM2 accumulated in registers. Writes eo rows.
// ---------------------------------------------------------------------------
__global__ void __launch_bounds__(256)
moe_ffn_kernel(const float* __restrict__ h2, const float* __restrict__ w1,
               const float* __restrict__ w2, const int* __restrict__ slot_tok,
               float* __restrict__ eo) {
  __shared__ float As[16][E_DIM + 4];  // 16 x 516 (conflict-free frag reads)
  __shared__ float Hs[16][256 + 4];    // 16 x 260
  const int tid = threadIdx.x;
  const int lane = tid & 31, wave = tid >> 5;
  const int lm = lane & 15, hi = lane >> 4;
  const int e = blockIdx.x >> 9;            // / 512
  const int b0 = (blockIdx.x & 511) << 2;   // * 4 groups

  for (int idx = tid; idx < 16 * E_DIM; idx += 256) {
    const int r = idx >> 9, col = idx & (E_DIM - 1);
    const int b = b0 + (r >> 2), c = r & 3;
    const int tok = slot_tok[((size_t)e * L_DIM + b) * CAP + c];
    As[r][col] = (tok >= 0)
                     ? h2[((size_t)b * B_DIM + tok) * E_DIM + col]
                     : 0.f;
  }
  __syncthreads();

  v8f oacc[4];
#pragma unroll
  for (int i = 0; i < 4; ++i) oacc[i] = v8f_zero();

  const float* w1base = w1 + (size_t)e * E_DIM * FF_DIM;
  for (int hc = 0; hc < FF_DIM; hc += 256) {
    // GEMM1: wave computes hidden cols [wave*32, wave*32+32)
    v8f hacc[2] = {v8f_zero(), v8f_zero()};
    for (int k = 0; k < E_DIM; k += 4) {
      v2f a = *(const v2f*)(&As[lm][k + 2 * hi]);
      const float* brow0 =
          w1base + (size_t)(k + 2 * hi) * FF_DIM + hc + wave * 32;
      const float* brow1 = brow0 + FF_DIM;
#pragma unroll
      for (int nt = 0; nt < 2; ++nt) {
        v2f bb;
        bb.x = brow0[nt * 16 + lm];
        bb.y = brow1[nt * 16 + lm];
        hacc[nt] = wmma_f32(a, bb, hacc[nt]);
      }
    }
    __syncthreads();  // prior GEMM2 done with Hs
#pragma unroll
    for (int nt = 0; nt < 2; ++nt)
#pragma unroll
      for (int r = 0; r < 8; ++r)
        Hs[r + 8 * hi][wave * 32 + nt * 16 + lm] = fmaxf(hacc[nt][r], 0.f);
    __syncthreads();
    // GEMM2: wave owns out cols [wave*64, wave*64+64), K = 256 chunk
    const float* w2base =
        w2 + (size_t)e * FF_DIM * E_DIM + (size_t)hc * E_DIM + wave * 64;
    for (int k = 0; k < 256; k += 4) {
      v2f a = *(const v2f*)(&Hs[lm][k + 2 * hi]);
      const float* brow0 = w2base + (size_t)(k + 2 * hi) * E_DIM;
      const float* brow1 = brow0 + E_DIM;
#pragma unroll
      for (int nt = 0; nt < 4; ++nt) {
        v2f bb;
        bb.x = brow0[nt * 16 + lm];
        bb.y = brow1[nt * 16 + lm];
        oacc[nt] = wmma_f32(a, bb, oacc[nt]);
      }
    }
  }
#pragma unroll
  for (int nt = 0; nt < 4; ++nt)
#pragma unroll
    for (int r = 0; r < 8; ++r) {
      const int row = r + 8 * hi;
      const int b = b0 + (row >> 2), c = row & 3;
      eo[(((size_t)e * L_DIM + b) * CAP + c) * E_DIM + wave * 64 + nt * 16 +
         lm] = oacc[nt][r];
    }
}

// ---------------------------------------------------------------------------
// Combine: out += g1*eo[e1,b,c1,:] + g2*eo[e2,b,c2,:] (deterministic gather).
// ---------------------------------------------------------------------------
__global__ void __launch_bounds__(256)
combine_kernel(const float* __restrict__ eo, const int* __restrict__ tok_e,
               const int* __restrict__ tok_c, const float* __restrict__ tok_g,
               float* __restrict__ out) {
  const int idx = blockIdx.x * 256 + threadIdx.x;
  if (idx >= TOKENS * E_DIM) return;
  const int t = idx >> 9, col = idx & (E_DIM - 1);
  const int b = t >> 2;
  float v = out[idx];
  {
    const float g = tok_g[t];
    if (g != 0.f) {
      const int e = tok_e[t], c = tok_c[t];
      v += g * eo[(((size_t)e * L_DIM + b) * CAP + c) * E_DIM + col];
    }
  }
  {
    const float g = tok_g[TOKENS + t];
    if (g != 0.f) {
      const int e = tok_e[TOKENS + t], c = tok_c[TOKENS + t];
      v += g * eo[(((size_t)e * L_DIM + b) * CAP + c) * E_DIM + col];
    }
  }
  out[idx] = v;
}

// ---------------------------------------------------------------------------
// Host-side orchestration. ws usage ~186 MB.
// ---------------------------------------------------------------------------
extern "C" void kernel_launch(void* const* d_in, const int* in_sizes, int n_in,
                              void* d_out, int out_size, void* d_ws,
                              size_t ws_size, hipStream_t stream) {
  (void)in_sizes; (void)n_in; (void)out_size; (void)ws_size;
  const float* x     = (const float*)d_in[0];
  const float* ln1w  = (const float*)d_in[1];
  const float* ln1b  = (const float*)d_in[2];
  const float* ln2w  = (const float*)d_in[3];
  const float* ln2b  = (const float*)d_in[4];
  const float* wi    = (const float*)d_in[5];   // (3E, E)
  const float* bi    = (const float*)d_in[6];   // (3E,)
  const float* wo    = (const float*)d_in[7];   // (E, E)
  const float* bo    = (const float*)d_in[8];   // (E,)
  const float* wg    = (const float*)d_in[9];   // (E, NE)
  const float* w1    = (const float*)d_in[10];  // (NE, E, FF)
  const float* w2    = (const float*)d_in[11];  // (NE, FF, E)
  float* out = (float*)d_out;

  float* ws = (float*)d_ws;
  size_t off = 0;
  float* h1    = ws + off; off += (size_t)TOKENS * E_DIM;      //  16.8 MB
  float* qkvb  = ws + off; off += (size_t)TOKENS * QKV_DIM;    //  50.3 MB
  float* attnO = ws + off; off += (size_t)TOKENS * E_DIM;      //  16.8 MB
  float* h2    = ws + off; off += (size_t)TOKENS * E_DIM;      //  16.8 MB
  float* eo    = ws + off; off += (size_t)NE_DIM * L_DIM * CAP * E_DIM; // 83.9 MB
  int*   slot_tok  = (int*)(ws + off);   off += (size_t)NE_DIM * L_DIM * CAP;
  float* slot_gate = ws + off;           off += (size_t)NE_DIM * L_DIM * CAP;
  int*   tok_e = (int*)(ws + off);       off += (size_t)2 * TOKENS;
  int*   tok_c = (int*)(ws + off);       off += (size_t)2 * TOKENS;
  float* tok_g = ws + off;               off += (size_t)2 * TOKENS;

  // 1. h1 = LN1(x)
  ln_kernel<<<TOKENS / 8, 256, 0, stream>>>(x, ln1w, ln1b, h1, TOKENS);
  // 2. qkv = h1 @ wi^T + bi   (16x64 strips: 512*24 waves)
  gemm_tn_wmma<<<(TOKENS / 16) * (QKV_DIM / 64) / 8, 256, 0, stream>>>(
      h1, wi, bi, nullptr, qkvb, TOKENS, QKV_DIM, E_DIM);
  // 3. attnO = softmax(QK^T/sqrt(hd)) V
  attn_kernel<<<B_DIM * H_DIM * (L_DIM / 16), 32, 0, stream>>>(qkvb, attnO);
  // 4. d_out = x + attnO @ wo^T + bo   (= x2)
  gemm_tn_wmma<<<(TOKENS / 16) * (E_DIM / 64) / 8, 256, 0, stream>>>(
      attnO, wo, bo, x, out, TOKENS, E_DIM, E_DIM);
  // 5. h2 = LN2(x2)
  ln_kernel<<<TOKENS / 8, 256, 0, stream>>>(out, ln2w, ln2b, h2, TOKENS);
  // 6. top-2 gating (threefry RNG, capacity)
  gating_kernel<<<L_DIM, 32, 0, stream>>>(h2, wg, slot_tok, slot_gate, tok_e,
                                          tok_c, tok_g);
  // 7. expert FFN over all (e, b, c) slots
  moe_ffn_kernel<<<NE_DIM * (L_DIM / 4), 256, 0, stream>>>(h2, w1, w2,
                                                           slot_tok, eo);
  // 8. d_out += combine gates * expert outputs
  combine_kernel<<<(TOKENS * E_DIM) / 256, 256, 0, stream>>>(eo, tok_e, tok_c,
                                                             tok_g, out);
}